// MPT_module_42202348650597
// MI455X (gfx1250) — compile-verified
//
#include <hip/hip_runtime.h>
#include <hip/hip_bf16.h>

typedef __attribute__((ext_vector_type(2))) float v2f;
typedef __attribute__((ext_vector_type(8))) float v8f;

#define EMA_EPS 1e-6f

// D = A(16x4) * B(4x16) + C, fp32 WMMA.
// Layout (ISA 7.12.2): A: lane m=0..15 rows; vgpr0/1 = K(2*half), K(2*half+1).
//                      B: lane n=0..15 cols; vgpr0/1 = K(2*half), K(2*half+1).
//                      C/D: d[j] = D[j + 8*half][lane&15].
__device__ __forceinline__ v8f wmma4(v2f a, v2f b, v8f c) {
  return __builtin_amdgcn_wmma_f32_16x16x4_f32(false, a, false, b, (short)0, c,
                                               false, false);
}

__device__ __forceinline__ v8f vzero8() {
  v8f r;
#pragma unroll
  for (int j = 0; j < 8; ++j) r[j] = 0.0f;
  return r;
}

// ---------------------------------------------------------------- utilities
__global__ void k_mu_init(const float* __restrict__ mu0, float* __restrict__ muA) {
  int i = blockIdx.x * blockDim.x + threadIdx.x;
  if (i < 4 * 256 * 128) muA[i] = mu0[i & 32767];  // broadcast over batch
}

__global__ void k_zero(float* __restrict__ p, int nel) {
  int i = blockIdx.x * blockDim.x + threadIdx.x;
  if (i < nel) p[i] = 0.0f;
}

// ------------------------------------------------- EM step 1: z = softmax(xf^T mu)
// grid: 4 batches * 256 pixel-tiles, block: 32 (one wave).
// Wave computes a [16 x 128] block of z with 8 WMMA accumulators, does the
// k-softmax in registers (shuffle across the 16-lane half-groups), writes z
// and atomically accumulates the per-column sums.
__global__ void k_em_z(const float* __restrict__ xf, const float* __restrict__ mu,
                       float* __restrict__ z, float* __restrict__ colsum) {
  const int L = 4096, C = 256, K = 128;
  int b = blockIdx.x >> 8;
  int l0 = (blockIdx.x & 255) * 16;
  int lane = threadIdx.x & 31;
  int half = lane >> 4, l16 = lane & 15;

  const float* X = xf + b * C * L;  // X[ch][l]
  const float* M = mu + b * C * K;  // M[ch][kk]

  v8f acc[8];
#pragma unroll
  for (int t = 0; t < 8; ++t) acc[t] = vzero8();

  for (int kc = 0; kc < C; kc += 4) {
    int ch = kc + 2 * half;
    v2f a;
    a[0] = X[ch * L + l0 + l16];        // A[m][ch] = xf[ch][l0+m]
    a[1] = X[(ch + 1) * L + l0 + l16];
#pragma unroll
    for (int t = 0; t < 8; ++t) {
      v2f bb;
      bb[0] = M[ch * K + t * 16 + l16];
      bb[1] = M[(ch + 1) * K + t * 16 + l16];
      acc[t] = wmma4(a, bb, acc[t]);
    }
  }
  // softmax over k=128: row m = j + 8*half; columns live at (reg t, lane l16).
#pragma unroll
  for (int j = 0; j < 8; ++j) {
    float mx = acc[0][j];
#pragma unroll
    for (int t = 1; t < 8; ++t) mx = fmaxf(mx, acc[t][j]);
    for (int s = 1; s < 16; s <<= 1) mx = fmaxf(mx, __shfl_xor(mx, s, 32));
    float sm = 0.0f;
#pragma unroll
    for (int t = 0; t < 8; ++t) {
      float e = __expf(acc[t][j] - mx);
      acc[t][j] = e;
      sm += e;
    }
    for (int s = 1; s < 16; s <<= 1) sm += __shfl_xor(sm, s, 32);
    float inv = 1.0f / sm;
#pragma unroll
    for (int t = 0; t < 8; ++t) acc[t][j] *= inv;
  }
  float* Zb = z + b * L * K;
#pragma unroll
  for (int t = 0; t < 8; ++t) {
    float cs = 0.0f;
#pragma unroll
    for (int j = 0; j < 8; ++j) {
      int row = j + 8 * half;
      Zb[(l0 + row) * K + t * 16 + l16] = acc[t][j];
      cs += acc[t][j];
    }
    atomicAdd(&colsum[b * K + t * 16 + l16], cs);
  }
}

// --------------------------------- EM step 2: mu_unnorm = xf @ (z / (eps+colsum))
// grid: 4 batches * 16 ch-tiles, block 128 (4 waves). xf tile staged via LDS
// (stride-65 padding, conflict free); each wave covers 32 k-columns.
__global__ void k_em_mu(const float* __restrict__ xf, const float* __restrict__ z,
                        const float* __restrict__ colsum, float* __restrict__ muB) {
  __shared__ float As[16][65];
  const int L = 4096, C = 256, K = 128;
  int b = blockIdx.x >> 4;
  int ch0 = (blockIdx.x & 15) * 16;
  int tid = threadIdx.x;
  int lane = tid & 31, wave = tid >> 5;
  int half = lane >> 4, l16 = lane & 15;
  int mRow = tid >> 3, cOff = (tid & 7) * 8;

  const float* X = xf + b * C * L;
  const float* Z = z + b * L * K;

  v8f acc0 = vzero8(), acc1 = vzero8();
  int n0 = wave * 32;

  for (int lc = 0; lc < L; lc += 64) {
    __syncthreads();
#pragma unroll
    for (int u = 0; u < 8; ++u)
      As[mRow][cOff + u] = X[(ch0 + mRow) * L + lc + cOff + u];
    __syncthreads();
#pragma unroll
    for (int kc = 0; kc < 64; kc += 4) {
      int k0 = kc + 2 * half;
      v2f a;
      a[0] = As[l16][k0];
      a[1] = As[l16][k0 + 1];
      int lr = lc + k0;
      v2f b0;
      b0[0] = Z[lr * K + n0 + l16];
      b0[1] = Z[(lr + 1) * K + n0 + l16];
      acc0 = wmma4(a, b0, acc0);
      v2f b1;
      b1[0] = Z[lr * K + n0 + 16 + l16];
      b1[1] = Z[(lr + 1) * K + n0 + 16 + l16];
      acc1 = wmma4(a, b1, acc1);
    }
  }
  float* MB = muB + b * C * K;
  {
    int n = n0 + l16;
    float ics = 1.0f / (EMA_EPS + colsum[b * K + n]);
#pragma unroll
    for (int j = 0; j < 8; ++j) MB[(ch0 + j + 8 * half) * K + n] = acc0[j] * ics;
  }
  {
    int n = n0 + 16 + l16;
    float ics = 1.0f / (EMA_EPS + colsum[b * K + n]);
#pragma unroll
    for (int j = 0; j < 8; ++j) MB[(ch0 + j + 8 * half) * K + n] = acc1[j] * ics;
  }
}

// -------------------------------------------- EM step 3: l2 norm over channel dim
__global__ void k_l2norm(const float* __restrict__ muB, float* __restrict__ muA) {
  __shared__ float red[256];
  int col = blockIdx.x;  // b*128 + kk
  int b = col >> 7, kk = col & 127;
  const float* src = muB + b * 256 * 128;
  float v = src[threadIdx.x * 128 + kk];
  red[threadIdx.x] = v * v;
  __syncthreads();
  for (int s = 128; s > 0; s >>= 1) {
    if ((int)threadIdx.x < s) red[threadIdx.x] += red[threadIdx.x + s];
    __syncthreads();
  }
  float inv = 1.0f / (EMA_EPS + sqrtf(red[0]));
  muA[b * 256 * 128 + threadIdx.x * 128 + kk] = v * inv;
}

// ---------------- transpose mu -> muT[b][k][c] and copy final mu into d_out tail
__global__ void k_mu_t(const float* __restrict__ muA, float* __restrict__ muT,
                       float* __restrict__ mu_out) {
  int i = blockIdx.x * blockDim.x + threadIdx.x;
  if (i >= 4 * 256 * 128) return;
  int b = i >> 15, r = i & 32767, kk = r >> 8, ch = r & 255;
  muT[(b << 15) + (kk << 8) + ch] = muA[(b << 15) + (ch << 7) + kk];
  mu_out[i] = muA[i];
}

// ---------------- querys NCHW -> cat NHWC (channels 0..255), LDS tiled transpose
__global__ void k_q_nhwc(const float* __restrict__ q, float* __restrict__ cat) {
  __shared__ float tile[32][33];
  int t = blockIdx.x;
  int img = t >> 10;  // 1024 tiles per image (8 ch-tiles * 128 p-tiles)
  int r = t & 1023;
  int ch0 = (r >> 7) * 32;
  int p0 = (r & 127) * 32;
  int tx = threadIdx.x & 31, ty = threadIdx.x >> 5;  // 32x8 threads
  const float* src = q + (size_t)img * 256 * 4096;
#pragma unroll
  for (int i = 0; i < 32; i += 8)
    tile[ty + i][tx] = src[(ch0 + ty + i) * 4096 + p0 + tx];
  __syncthreads();
  float* dst = cat + (size_t)img * 4096 * 512;
#pragma unroll
  for (int i = 0; i < 32; i += 8)
    dst[(size_t)(p0 + ty + i) * 512 + ch0 + tx] = tile[tx][ty + i];
}

// -------- fused attention: q_z = softmax(q^T mu); rec = mu q_z^T -> cat[...,256:]
// grid: 16 imgs * 256 pixel-tiles, block 128 (4 waves).
__global__ void k_attn(const float* __restrict__ querys, const float* __restrict__ mu,
                       const float* __restrict__ muT, float* __restrict__ cat) {
  __shared__ float zs[16][129];
  const int HW = 4096, C = 256, K = 128;
  int img = blockIdx.x >> 8;
  int p0 = (blockIdx.x & 255) * 16;
  int bb_ = img >> 2;  // batch index
  int tid = threadIdx.x;
  int lane = tid & 31, wave = tid >> 5;
  int half = lane >> 4, l16 = lane & 15;

  const float* Q = querys + (size_t)img * C * HW;
  const float* M = mu + bb_ * C * K;

  // phase 1: z = q^T mu ; wave covers k-columns [32*wave, 32*wave+32)
  v8f acc0 = vzero8(), acc1 = vzero8();
  int n0 = wave * 32;
  for (int kc = 0; kc < C; kc += 4) {
    int ch = kc + 2 * half;
    v2f a;
    a[0] = Q[ch * HW + p0 + l16];
    a[1] = Q[(ch + 1) * HW + p0 + l16];
    v2f b0;
    b0[0] = M[ch * K + n0 + l16];
    b0[1] = M[(ch + 1) * K + n0 + l16];
    acc0 = wmma4(a, b0, acc0);
    v2f b1;
    b1[0] = M[ch * K + n0 + 16 + l16];
    b1[1] = M[(ch + 1) * K + n0 + 16 + l16];
    acc1 = wmma4(a, b1, acc1);
  }
#pragma unroll
  for (int j = 0; j < 8; ++j) {
    int row = j + 8 * half;
    zs[row][n0 + l16] = acc0[j];
    zs[row][n0 + 16 + l16] = acc1[j];
  }
  __syncthreads();
  // softmax over k=128: 8 threads per pixel row
  {
    int row = tid >> 3, s8 = tid & 7;
    float mx = -3.0e38f;
#pragma unroll
    for (int i = 0; i < 16; ++i) mx = fmaxf(mx, zs[row][s8 + 8 * i]);
    mx = fmaxf(mx, __shfl_xor(mx, 1, 32));
    mx = fmaxf(mx, __shfl_xor(mx, 2, 32));
    mx = fmaxf(mx, __shfl_xor(mx, 4, 32));
    float ev[16];
    float sm = 0.0f;
#pragma unroll
    for (int i = 0; i < 16; ++i) {
      ev[i] = __expf(zs[row][s8 + 8 * i] - mx);
      sm += ev[i];
    }
    sm += __shfl_xor(sm, 1, 32);
    sm += __shfl_xor(sm, 2, 32);
    sm += __shfl_xor(sm, 4, 32);
    float inv = 1.0f / sm;
#pragma unroll
    for (int i = 0; i < 16; ++i) zs[row][s8 + 8 * i] = ev[i] * inv;
  }
  __syncthreads();
  // phase 2: rec[16 x 256] = z_[16 x 128] @ muT[128 x 256]; wave covers 64 ch
  const float* MT = muT + bb_ * K * C;
  v8f r0 = vzero8(), r1 = vzero8(), r2 = vzero8(), r3 = vzero8();
  int chw = wave * 64;
  for (int kc = 0; kc < K; kc += 4) {
    int k0 = kc + 2 * half;
    v2f a;
    a[0] = zs[l16][k0];
    a[1] = zs[l16][k0 + 1];
    v2f bv;
    bv[0] = MT[k0 * C + chw + l16];
    bv[1] = MT[(k0 + 1) * C + chw + l16];
    r0 = wmma4(a, bv, r0);
    bv[0] = MT[k0 * C + chw + 16 + l16];
    bv[1] = MT[(k0 + 1) * C + chw + 16 + l16];
    r1 = wmma4(a, bv, r1);
    bv[0] = MT[k0 * C + chw + 32 + l16];
    bv[1] = MT[(k0 + 1) * C + chw + 32 + l16];
    r2 = wmma4(a, bv, r2);
    bv[0] = MT[k0 * C + chw + 48 + l16];
    bv[1] = MT[(k0 + 1) * C + chw + 48 + l16];
    r3 = wmma4(a, bv, r3);
  }
  float* dst = cat + (size_t)img * HW * 512;
#pragma unroll
  for (int j = 0; j < 8; ++j) {
    int p = p0 + j + 8 * half;
    float* row = dst + (size_t)p * 512 + 256 + chw + l16;
    row[0] = r0[j];
    row[16] = r1[j];
    row[32] = r2[j];
    row[48] = r3[j];
  }
}

// -------------------- weight repack: conv_w[oc][ic][kh][kw] -> wT[tap][ic][oc]
__global__ void k_w_t(const float* __restrict__ w, float* __restrict__ wT) {
  int i = blockIdx.x * blockDim.x + threadIdx.x;
  if (i >= 9 * 512 * 256) return;
  int tap = i / (512 * 256);
  int r = i % (512 * 256);
  int ic = r >> 8, oc = r & 255;
  wT[i] = w[oc * 512 * 9 + ic * 9 + tap];
}

// --------------------------------- conv 3x3 SAME as implicit GEMM (M=16 pixels)
// grid: 16 imgs * 64 rows * 4 x-tiles, block 128 (4 waves). A tile staged in LDS
// and shared across waves; B loads contiguous from repacked weights.
__global__ void k_conv(const float* __restrict__ cat, const float* __restrict__ wT,
                       const float* __restrict__ bias, float* __restrict__ out) {
  __shared__ float As[16][65];
  int blk = blockIdx.x;
  int img = blk >> 8;
  int r = blk & 255;
  int y = r >> 2;
  int x0 = (r & 3) * 16;
  int tid = threadIdx.x;
  int lane = tid & 31, wave = tid >> 5;
  int half = lane >> 4, l16 = lane & 15;
  int mRow = tid >> 3, cOff = (tid & 7) * 8;

  const float* src = cat + (size_t)img * 4096 * 512;
  v8f acc[4];
#pragma unroll
  for (int t = 0; t < 4; ++t) acc[t] = vzero8();

  for (int tap = 0; tap < 9; ++tap) {
    int dy = tap / 3 - 1, dx = tap % 3 - 1;
    int yy = y + dy;
    int xx = x0 + mRow + dx;
    bool ok = (yy >= 0) && (yy < 64) && (xx >= 0) && (xx < 64);
    const float* srow = src + (size_t)(yy * 64 + xx) * 512;
    for (int cc0 = 0; cc0 < 512; cc0 += 64) {
      __syncthreads();
      if (ok) {
#pragma unroll
        for (int u = 0; u < 8; ++u) As[mRow][cOff + u] = srow[cc0 + cOff + u];
      } else {
#pragma unroll
        for (int u = 0; u < 8; ++u) As[mRow][cOff + u] = 0.0f;
      }
      __syncthreads();
      const float* W = wT + ((size_t)tap * 512 + cc0) * 256;
#pragma unroll
      for (int kc = 0; kc < 64; kc += 4) {
        int k0 = kc + 2 * half;
        v2f a;
        a[0] = As[l16][k0];
        a[1] = As[l16][k0 + 1];
#pragma unroll
        for (int t = 0; t < 4; ++t) {
          int oc0 = wave * 64 + t * 16;
          v2f bb;
          bb[0] = W[(size_t)k0 * 256 + oc0 + l16];
          bb[1] = W[(size_t)(k0 + 1) * 256 + oc0 + l16];
          acc[t] = wmma4(a, bb, acc[t]);
        }
      }
    }
  }
  float* O = out + (size_t)img * 256 * 4096;
#pragma unroll
  for (int t = 0; t < 4; ++t) {
    int oc = wave * 64 + t * 16 + l16;
    float bv = bias[oc];
#pragma unroll
    for (int j = 0; j < 8; ++j) {
      int m = j + 8 * half;
      O[(size_t)oc * 4096 + y * 64 + x0 + m] = acc[t][j] + bv;
    }
  }
}

// ---------------------------------------------------------------------- launch
extern "C" void kernel_launch(void* const* d_in, const int* in_sizes, int n_in,
                              void* d_out, int out_size, void* d_ws, size_t ws_size,
                              hipStream_t stream) {
  const float* x = (const float*)d_in[0];       // [4,256,64,64]
  const float* querys = (const float*)d_in[1];  // [16,256,64,64]
  const float* mu0 = (const float*)d_in[2];     // [1,256,128]
  const float* conv_w = (const float*)d_in[3];  // [256,512,3,3]
  const float* conv_b = (const float*)d_in[4];  // [256]
  float* out = (float*)d_out;                   // out(16.7M) then mu(131072)

  float* ws = (float*)d_ws;
  float* muA = ws;                    // 131072
  float* muB = muA + 131072;          // 131072
  float* muT = muB + 131072;          // 131072
  float* colsum = muT + 131072;       // 512
  float* z = colsum + 512;            // 4*4096*128 = 2097152
  float* cat = z + 2097152;           // 16*4096*512 = 33554432 (NHWC concat)
  float* wT = cat + 33554432;         // 9*512*256 = 1179648

  k_mu_init<<<512, 256, 0, stream>>>(mu0, muA);
  for (int it = 0; it < 5; ++it) {
    k_zero<<<1, 512, 0, stream>>>(colsum, 512);
    k_em_z<<<1024, 32, 0, stream>>>(x, muA, z, colsum);
    k_em_mu<<<64, 128, 0, stream>>>(x, z, colsum, muB);
    k_l2norm<<<512, 256, 0, stream>>>(muB, muA);
  }
  k_mu_t<<<512, 256, 0, stream>>>(muA, muT, out + 16777216);
  k_q_nhwc<<<16384, 256, 0, stream>>>(querys, cat);
  k_attn<<<4096, 128, 0, stream>>>(querys, muA, muT, cat);
  k_w_t<<<(9 * 512 * 256 + 255) / 256, 256, 0, stream>>>(conv_w, wT);
  k_conv<<<4096, 128, 0, stream>>>(cat, wT, conv_b, out);
}